// ExpertGNN_36782099923592
// MI455X (gfx1250) — compile-verified
//
#include <hip/hip_runtime.h>
#include <math.h>

#define BATCH  64
#define NNODES 4096
#define GRIDW  64
#define HID    128
#define OUTC   64
#define LN_EPS 1e-5f

typedef float v2f __attribute__((ext_vector_type(2)));
typedef float v8f __attribute__((ext_vector_type(8)));

__device__ __forceinline__ float gelu_exact(float v) {
    return 0.5f * v * (1.0f + erff(v * 0.70710678118654752f));
}

// -------------------------------------------------------------------------
// h0 = gelu(LN(x_nodes @ W0 + b0))   [B*N rows, 128 channels]
// One block per row, one thread per channel. K=3 so plain VALU is optimal.
// -------------------------------------------------------------------------
__global__ __launch_bounds__(128)
void embed_kernel(const float* __restrict__ x,  const float* __restrict__ W0,
                  const float* __restrict__ b0, const float* __restrict__ g0,
                  const float* __restrict__ be0, float* __restrict__ h0) {
    const int r  = blockIdx.x;            // b*N + n
    const int b  = r >> 12;
    const int n  = r & (NNODES - 1);
    const int ch = threadIdx.x;

    const float x0 = x[(b * 3 + 0) * NNODES + n];
    const float x1 = x[(b * 3 + 1) * NNODES + n];
    const float x2 = x[(b * 3 + 2) * NNODES + n];
    float t = x0 * W0[ch] + x1 * W0[HID + ch] + x2 * W0[2 * HID + ch] + b0[ch];

    __shared__ float red[128];
    red[ch] = t; __syncthreads();
    for (int s = 64; s > 0; s >>= 1) { if (ch < s) red[ch] += red[ch + s]; __syncthreads(); }
    const float mean = red[0] * (1.0f / 128.0f);
    __syncthreads();
    const float d = t - mean;
    red[ch] = d * d; __syncthreads();
    for (int s = 64; s > 0; s >>= 1) { if (ch < s) red[ch] += red[ch + s]; __syncthreads(); }
    const float var = red[0] * (1.0f / 128.0f);

    const float yv = d * rsqrtf(var + LN_EPS) * g0[ch] + be0[ch];
    h0[(size_t)r * HID + ch] = gelu_exact(yv);
}

// -------------------------------------------------------------------------
// Fused GNN layer: msg = adj @ h (49-point normalized stencil, sparse),
// then h_out = gelu(LN(msg @ W, g, be)) via V_WMMA_F32_16X16X4_F32.
// Block = 128 threads (4 waves), owns 16 consecutive rows (one batch).
// -------------------------------------------------------------------------
__global__ __launch_bounds__(128)
void layer_kernel(const float* __restrict__ hin, const float* __restrict__ adj,
                  const float* __restrict__ W,   const float* __restrict__ g,
                  const float* __restrict__ be,  float* __restrict__ hout) {
    const int APAD = 132;                        // 132 % 64 == 4 -> conflict-free
    __shared__ float msg_s[16 * APAD];
    __shared__ float out_s[16 * APAD];

    const int r0   = blockIdx.x * 16;
    const int b    = r0 >> 12;
    const int n0   = r0 & (NNODES - 1);
    const int y0   = n0 >> 6;
    const int xx0b = n0 & 63;                    // n0 % 16 == 0 -> no row wrap over 16 nodes
    const int tid  = threadIdx.x;
    const int wave = tid >> 5;
    const int lane = tid & 31;
    const int c4   = lane << 2;                  // 4 channels per lane

    // ---- sparse message passing: exact support of (dist <= 4) adjacency ----
    for (int j = 0; j < 4; ++j) {
        const int i  = wave * 4 + j;             // node within tile (wave-uniform)
        const int n  = n0 + i;
        const int xc = xx0b + i;
        float4 acc = make_float4(0.f, 0.f, 0.f, 0.f);
        for (int dy = -4; dy <= 4; ++dy) {
            const int yy = y0 + dy;
            if (yy < 0 || yy >= GRIDW) continue;
            const int lim = 16 - dy * dy;
            for (int dx = -4; dx <= 4; ++dx) {
                if (dx * dx > lim) continue;
                const int xx = xc + dx;
                if (xx < 0 || xx >= GRIDW) continue;
                const int m = (yy << 6) + xx;
                const float a = adj[n * NNODES + m];            // wave-uniform -> s_load
                const float4 hv = *reinterpret_cast<const float4*>(
                    &hin[((size_t)(b * NNODES + m)) * HID + c4]);
                acc.x += a * hv.x; acc.y += a * hv.y;
                acc.z += a * hv.z; acc.w += a * hv.w;
            }
        }
        *reinterpret_cast<float4*>(&msg_s[i * APAD + c4]) = acc;
    }
    __syncthreads();

    // ---- 16x128 = (16x128) @ (128x128) with fp32 WMMA, 2 col-tiles per wave ----
    const int half = lane >> 4;                  // 0: lanes 0-15, 1: lanes 16-31
    const int l16  = lane & 15;
    const int c0   = wave * 32;
    v8f acc0 = {}; v8f acc1 = {};
    #pragma unroll 4
    for (int k = 0; k < HID; k += 4) {
        const int ka = k + half * 2;
        v2f a;                                    // A 16x4 fp32 fragment
        a.x = msg_s[l16 * APAD + ka];
        a.y = msg_s[l16 * APAD + ka + 1];
        v2f bf0, bf1;                             // B 4x16 fp32 fragments
        bf0.x = W[ka * HID + c0 + l16];
        bf0.y = W[(ka + 1) * HID + c0 + l16];
        bf1.x = W[ka * HID + c0 + 16 + l16];
        bf1.y = W[(ka + 1) * HID + c0 + 16 + l16];
        acc0 = __builtin_amdgcn_wmma_f32_16x16x4_f32(false, a, false, bf0,
                                                     (short)0, acc0, false, false);
        acc1 = __builtin_amdgcn_wmma_f32_16x16x4_f32(false, a, false, bf1,
                                                     (short)0, acc1, false, false);
    }
    #pragma unroll
    for (int i = 0; i < 8; ++i) {
        const int row = i + half * 8;
        out_s[row * APAD + c0 + l16]      = acc0[i];
        out_s[row * APAD + c0 + 16 + l16] = acc1[i];
    }
    __syncthreads();

    // ---- LayerNorm + gelu, wave w handles rows 4w..4w+3 ----
    for (int j = 0; j < 4; ++j) {
        const int i = wave * 4 + j;
        const float4 v = *reinterpret_cast<const float4*>(&out_s[i * APAD + c4]);
        float ps = v.x + v.y + v.z + v.w;
        float pq = v.x * v.x + v.y * v.y + v.z * v.z + v.w * v.w;
        for (int m = 16; m > 0; m >>= 1) {
            ps += __shfl_xor(ps, m, 32);
            pq += __shfl_xor(pq, m, 32);
        }
        const float mean = ps * (1.0f / 128.0f);
        const float var  = pq * (1.0f / 128.0f) - mean * mean;
        const float inv  = rsqrtf(var + LN_EPS);
        float4 o;
        o.x = gelu_exact((v.x - mean) * inv * g[c4 + 0] + be[c4 + 0]);
        o.y = gelu_exact((v.y - mean) * inv * g[c4 + 1] + be[c4 + 1]);
        o.z = gelu_exact((v.z - mean) * inv * g[c4 + 2] + be[c4 + 2]);
        o.w = gelu_exact((v.w - mean) * inv * g[c4 + 3] + be[c4 + 3]);
        *reinterpret_cast<float4*>(&hout[((size_t)(r0 + i)) * HID + c4]) = o;
    }
}

// -------------------------------------------------------------------------
// out = LN((h3+h0) @ Wo + bo), transposed to [B, OUT_C, N]. WMMA again.
// -------------------------------------------------------------------------
__global__ __launch_bounds__(128)
void out_kernel(const float* __restrict__ h3, const float* __restrict__ h0,
                const float* __restrict__ Wo, const float* __restrict__ bo,
                const float* __restrict__ go, const float* __restrict__ beo,
                float* __restrict__ out) {
    const int APAD = 132;
    const int OPAD = 68;
    __shared__ float hs[16 * APAD];
    __shared__ float os[16 * OPAD];

    const int r0   = blockIdx.x * 16;
    const int b    = r0 >> 12;
    const int n0   = r0 & (NNODES - 1);
    const int tid  = threadIdx.x;
    const int wave = tid >> 5;
    const int lane = tid & 31;

    for (int idx = tid; idx < 16 * HID; idx += 128) {
        const int i = idx >> 7, ch = idx & 127;
        const size_t gi = (size_t)(r0 + i) * HID + ch;
        hs[i * APAD + ch] = h3[gi] + h0[gi];
    }
    __syncthreads();

    const int half = lane >> 4;
    const int l16  = lane & 15;
    const int c0   = wave * 16;                  // 4 waves cover 64 output cols
    v8f acc = {};
    #pragma unroll 4
    for (int k = 0; k < HID; k += 4) {
        const int ka = k + half * 2;
        v2f a;  a.x = hs[l16 * APAD + ka];       a.y = hs[l16 * APAD + ka + 1];
        v2f bf; bf.x = Wo[ka * OUTC + c0 + l16]; bf.y = Wo[(ka + 1) * OUTC + c0 + l16];
        acc = __builtin_amdgcn_wmma_f32_16x16x4_f32(false, a, false, bf,
                                                    (short)0, acc, false, false);
    }
    #pragma unroll
    for (int i = 0; i < 8; ++i) {
        const int row = i + half * 8;
        os[row * OPAD + c0 + l16] = acc[i] + bo[c0 + l16];
    }
    __syncthreads();

    // LN over 64 channels; wave handles rows 4w..4w+3, lane owns 2 channels
    const int c2 = lane * 2;
    for (int j = 0; j < 4; ++j) {
        const int i = wave * 4 + j;
        const float v0 = os[i * OPAD + c2];
        const float v1 = os[i * OPAD + c2 + 1];
        float ps = v0 + v1, pq = v0 * v0 + v1 * v1;
        for (int m = 16; m > 0; m >>= 1) {
            ps += __shfl_xor(ps, m, 32);
            pq += __shfl_xor(pq, m, 32);
        }
        const float mean = ps * (1.0f / 64.0f);
        const float var  = pq * (1.0f / 64.0f) - mean * mean;
        const float inv  = rsqrtf(var + LN_EPS);
        os[i * OPAD + c2]     = (v0 - mean) * inv * go[c2]     + beo[c2];
        os[i * OPAD + c2 + 1] = (v1 - mean) * inv * go[c2 + 1] + beo[c2 + 1];
    }
    __syncthreads();

    // transposed, coalesced-in-n store: out[b, c, n0+i]
    for (int idx = tid; idx < 16 * OUTC; idx += 128) {
        const int c = idx >> 4, i = idx & 15;
        out[((size_t)(b * OUTC + c)) * NNODES + n0 + i] = os[i * OPAD + c];
    }
}

extern "C" void kernel_launch(void* const* d_in, const int* in_sizes, int n_in,
                              void* d_out, int out_size, void* d_ws, size_t ws_size,
                              hipStream_t stream) {
    const float* x   = (const float*)d_in[0];
    const float* adj = (const float*)d_in[1];
    const float* W0  = (const float*)d_in[2];
    const float* b0  = (const float*)d_in[3];
    const float* g0  = (const float*)d_in[4];
    const float* be0 = (const float*)d_in[5];
    const float* W1  = (const float*)d_in[6];
    const float* g1  = (const float*)d_in[7];
    const float* be1 = (const float*)d_in[8];
    const float* W2  = (const float*)d_in[9];
    const float* g2  = (const float*)d_in[10];
    const float* be2 = (const float*)d_in[11];
    const float* W3  = (const float*)d_in[12];
    const float* g3  = (const float*)d_in[13];
    const float* be3 = (const float*)d_in[14];
    const float* Wo  = (const float*)d_in[15];
    const float* bo  = (const float*)d_in[16];
    const float* go  = (const float*)d_in[17];
    const float* beo = (const float*)d_in[18];
    float* out = (float*)d_out;
    (void)in_sizes; (void)n_in; (void)out_size; (void)ws_size;

    const size_t SZ = (size_t)BATCH * NNODES * HID;   // floats per activation buffer
    float* h0v = (float*)d_ws;
    float* hA  = h0v + SZ;
    float* hB  = hA + SZ;

    embed_kernel<<<BATCH * NNODES,      128, 0, stream>>>(x, W0, b0, g0, be0, h0v);
    layer_kernel<<<BATCH * NNODES / 16, 128, 0, stream>>>(h0v, adj, W1, g1, be1, hA);
    layer_kernel<<<BATCH * NNODES / 16, 128, 0, stream>>>(hA,  adj, W2, g2, be2, hB);
    layer_kernel<<<BATCH * NNODES / 16, 128, 0, stream>>>(hB,  adj, W3, g3, be3, hA);
    out_kernel  <<<BATCH * NNODES / 16, 128, 0, stream>>>(hA, h0v, Wo, bo, go, beo, out);
}